// MIND_loss_9096740733042
// MI455X (gfx1250) — compile-verified
//
#include <hip/hip_runtime.h>
#include <math.h>
#include <stdint.h>

typedef __attribute__((ext_vector_type(2))) float v2f;
typedef __attribute__((ext_vector_type(8))) float v8f;

#define TX 16
#define TY 16
#define TZ 8
#define N3 128
#define IMX (TX + 8)   // 24
#define IMY (TY + 8)   // 24
#define IMZ (TZ + 8)   // 16
#define DX (TX + 4)    // 20
#define DY (TY + 4)    // 20
#define DZ (TZ + 4)    // 12
#define NVOX (TX * TY * TZ)   // 2048
#define NTHREADS 256
#define NBLOCKS ((N3 / TX) * (N3 / TY) * (N3 / TZ))  // 1024

#define MODE_STATS 0
#define MODE_WRITE 1
#define MODE_LOSS  2

// CDNA5 async global->LDS path (ASYNCcnt-tracked), with safe fallback.
#if defined(__AMDGCN__) && __has_builtin(__builtin_amdgcn_global_load_async_to_lds_b32)
#define HAVE_ASYNC_LDS 1
typedef __attribute__((address_space(1))) int gint_as;   // global int*
typedef __attribute__((address_space(3))) int lint_as;   // LDS int*
#else
#define HAVE_ASYNC_LDS 0
#endif

__device__ __forceinline__ void wait_async_lds() {
#if HAVE_ASYNC_LDS
#if __has_builtin(__builtin_amdgcn_s_wait_asynccnt)
  __builtin_amdgcn_s_wait_asynccnt(0);
#else
  asm volatile("s_wait_asynccnt 0" ::: "memory");
#endif
#endif
}

__device__ __forceinline__ int clamp128(int v) { return v < 0 ? 0 : (v > 127 ? 127 : v); }

// Shift pairs (z,y,x) derived from the reference's one-hot 3x3x3 dilated kernels:
// pairs (x>y, dist^2==2) over six = {(0,1,1),(1,1,0),(1,0,1),(1,1,2),(2,1,1),(1,2,1)},
// shift = 2*(six-1).
__device__ const int SH1[12][3] = {
  {0,0,-2},{0,-2,0},{0,-2,0},{0,0,2},{0,0,2},{2,0,0},
  {2,0,0},{2,0,0},{0,2,0},{0,2,0},{0,2,0},{0,2,0}};
__device__ const int SH2[12][3] = {
  {-2,0,0},{-2,0,0},{0,0,-2},{-2,0,0},{0,-2,0},{0,0,-2},
  {0,-2,0},{0,0,2},{-2,0,0},{0,0,-2},{0,0,2},{2,0,0}};

// Stage one halo tile into LDS. With the async path each lane issues a
// GLOBAL_LOAD_ASYNC_TO_LDS_B32 (per-lane global addr -> per-lane LDS addr),
// so the clamped replicate-halo gather goes straight to LDS without a VGPR
// round trip. Caller must wait_async_lds() + __syncthreads() before use.
__device__ __forceinline__ void load_tile(float* simg, const float* img,
                                          int gx0, int gy0, int gz0) {
  for (int i = threadIdx.x; i < IMZ * IMY * IMX; i += NTHREADS) {
    int lz = i / (IMY * IMX);
    int r  = i % (IMY * IMX);
    int ly = r / IMX, lx = r % IMX;
    int gz = clamp128(gz0 - 4 + lz);
    int gy = clamp128(gy0 - 4 + ly);
    int gx = clamp128(gx0 - 4 + lx);
    const float* gp = &img[(gz * N3 + gy) * N3 + gx];
#if HAVE_ASYNC_LDS
    // Address-space conversion via integers: generic global ptr value == AS1
    // address; low 32 bits of a generic LDS ptr == hardware LDS offset (AS3).
    __builtin_amdgcn_global_load_async_to_lds_b32(
        (gint_as*)(uintptr_t)gp,
        (lint_as*)(uint32_t)(uintptr_t)&simg[i], 0, 0);
#else
    simg[i] = *gp;
#endif
  }
}

// Computes ssd (box-filtered squared shift-difference) for channel c on one tile.
// bufA: DZ*DY*DX floats (diff2, later reused for t2). bufB: DZ*DY*TX floats (t1).
__device__ void box_channel(const float* simg, int c, int gx0, int gy0, int gz0,
                            float* bufA, float* bufB, int mode,
                            float* statMin, float* statSum,
                            float* sssd,
                            const float* minT, const float* mvT,
                            const float* minP, const float* mvP,
                            float tLo, float tHi, float pLo, float pHi,
                            float* lossAcc) {
  float* sdiff = bufA;
  float* st1   = bufB;
  float* st2   = bufA;  // reuse: diff2 dead after x-pass
  const int s1z = SH1[c][0], s1y = SH1[c][1], s1x = SH1[c][2];
  const int s2z = SH2[c][0], s2y = SH2[c][1], s2x = SH2[c][2];

  __syncthreads();  // previous stage (z-pass) done reading bufA/bufB/sssd

  // ---- diff2 stage (evaluated at replicate-clamped coordinates) ----
  for (int i = threadIdx.x; i < DZ * DY * DX; i += NTHREADS) {
    int dz = i / (DY * DX);
    int r  = i % (DY * DX);
    int dy = r / DX, dx = r % DX;
    int cz = clamp128(gz0 - 2 + dz);
    int cy = clamp128(gy0 - 2 + dy);
    int cx = clamp128(gx0 - 2 + dx);
    int az = clamp128(cz + s1z) - gz0 + 4;
    int ay = clamp128(cy + s1y) - gy0 + 4;
    int ax = clamp128(cx + s1x) - gx0 + 4;
    int bz = clamp128(cz + s2z) - gz0 + 4;
    int by = clamp128(cy + s2y) - gy0 + 4;
    int bx = clamp128(cx + s2x) - gx0 + 4;
    float d = simg[(az * IMY + ay) * IMX + ax] - simg[(bz * IMY + by) * IMX + bx];
    sdiff[i] = d * d;
  }
  __syncthreads();

  // ---- x-pass as banded matmul on the matrix pipe ----
  // Out(16 rows x 16 cols) = Data(16x20) x Band(20x16), Band[u][n] = (n<=u<=n+4).
  // 240 rows (DZ*DY) -> 15 groups of 16; K=20 split into 5 chained K=4 WMMAs.
  {
    const int lane = threadIdx.x & 31;
    const int wv   = threadIdx.x >> 5;   // wave id, 8 waves (wave-uniform loop)
    const int m    = lane & 15;          // A row / B column
    const int g    = lane >> 4;          // half-wave group
    for (int G = wv; G < (DZ * DY) / 16; G += NTHREADS / 32) {
      v8f acc = {0.f, 0.f, 0.f, 0.f, 0.f, 0.f, 0.f, 0.f};
      const float* rowp = &sdiff[(16 * G + m) * DX];
#pragma unroll
      for (int j = 0; j < 5; ++j) {
        const int u0 = 4 * j + 2 * g;    // A layout: vgpr0 holds K=2g, vgpr1 K=2g+1
        v2f A;
        A[0] = rowp[u0];
        A[1] = rowp[u0 + 1];
        v2f B;                            // band bits, same K mapping
        B[0] = (u0     >= m && u0     <= m + 4) ? 1.0f : 0.0f;
        B[1] = (u0 + 1 >= m && u0 + 1 <= m + 4) ? 1.0f : 0.0f;
        acc = __builtin_amdgcn_wmma_f32_16x16x4_f32(
            false, A, false, B, (short)0, acc, false, false);
      }
#pragma unroll
      for (int v = 0; v < 8; ++v)        // D layout: vgpr v -> row v + 8g, col m
        st1[(16 * G + v + 8 * g) * TX + m] = acc[v];
    }
  }
  __syncthreads();

  // ---- y-pass ----
  for (int i = threadIdx.x; i < DZ * TY * TX; i += NTHREADS) {
    int dz = i / (TY * TX);
    int r  = i % (TY * TX);
    int oy = r / TX, ox = r % TX;
    float s = 0.f;
#pragma unroll
    for (int k = 0; k < 5; ++k) s += st1[(dz * DY + oy + k) * TX + ox];
    st2[i] = s;  // layout (dz*TY+oy)*TX+ox
  }
  __syncthreads();

  // ---- z-pass + consume ----
  for (int i = threadIdx.x; i < NVOX; i += NTHREADS) {
    int oz = i / (TY * TX);
    int r  = i % (TY * TX);
    float s = 0.f;
#pragma unroll
    for (int k = 0; k < 5; ++k) s += st2[(oz + k) * TY * TX + r];
    float ssd = s * (1.0f / 125.0f);
    if (mode == MODE_STATS) {
      statMin[i] = fminf(statMin[i], ssd);
      statSum[i] += ssd;
    } else if (mode == MODE_WRITE) {
      sssd[i] = ssd;
    } else {  // MODE_LOSS: this image is P, sssd holds T's ssd for this channel
      float mindT = sssd[i] - minT[i];
      float vT = fminf(fmaxf(mvT[i], tLo), tHi);
      float eT = expf(-mindT / vT);
      float mindP = ssd - minP[i];
      float vP = fminf(fmaxf(mvP[i], pLo), pHi);
      float eP = expf(-mindP / vP);
      float dd = eT - eP;
      *lossAcc += dd * dd;
    }
  }
}

__device__ __forceinline__ void block_reduce_store(float acc, float* sred, float* dst) {
  sred[threadIdx.x] = acc;
  __syncthreads();
  for (int s = NTHREADS / 2; s > 0; s >>= 1) {
    if (threadIdx.x < (unsigned)s) sred[threadIdx.x] += sred[threadIdx.x + s];
    __syncthreads();
  }
  if (threadIdx.x == 0) *dst = sred[0];
}

// Phase 1: per-image, partial sums of mind_var (for global mean m).
__global__ __launch_bounds__(NTHREADS) void mind_phase1(const float* img, float* partials) {
  __shared__ float simg[IMZ * IMY * IMX];
  __shared__ float bufA[DZ * DY * DX];
  __shared__ float bufB[DZ * DY * TX];
  __shared__ float statMin[NVOX];
  __shared__ float statSum[NVOX];
  __shared__ float sred[NTHREADS];
  const int gx0 = blockIdx.x * TX, gy0 = blockIdx.y * TY, gz0 = blockIdx.z * TZ;

  load_tile(simg, img, gx0, gy0, gz0);
  for (int i = threadIdx.x; i < NVOX; i += NTHREADS) { statMin[i] = 3.4e38f; statSum[i] = 0.f; }
  wait_async_lds();

  for (int c = 0; c < 12; ++c)
    box_channel(simg, c, gx0, gy0, gz0, bufA, bufB, MODE_STATS,
                statMin, statSum, nullptr, nullptr, nullptr, nullptr, nullptr,
                0.f, 0.f, 0.f, 0.f, nullptr);
  __syncthreads();

  float acc = 0.f;
  for (int i = threadIdx.x; i < NVOX; i += NTHREADS)
    acc += statSum[i] * (1.0f / 12.0f) - statMin[i];

  const int bid = blockIdx.x + gridDim.x * (blockIdx.y + gridDim.y * blockIdx.z);
  block_reduce_store(acc, sred, &partials[bid]);
}

// Reduce partials -> m scalar for each image (blockIdx.x: 0 = true, 1 = pred).
__global__ __launch_bounds__(NTHREADS) void mind_reduce_m(const float* pT, const float* pP,
                                                          float* mT, float* mP) {
  __shared__ float sred[NTHREADS];
  const float* p = (blockIdx.x == 0) ? pT : pP;
  float* o = (blockIdx.x == 0) ? mT : mP;
  float acc = 0.f;
  for (int i = threadIdx.x; i < NBLOCKS; i += NTHREADS) acc += p[i];
  sred[threadIdx.x] = acc;
  __syncthreads();
  for (int s = NTHREADS / 2; s > 0; s >>= 1) {
    if (threadIdx.x < (unsigned)s) sred[threadIdx.x] += sred[threadIdx.x + s];
    __syncthreads();
  }
  if (threadIdx.x == 0) *o = sred[0] * (1.0f / 2097152.0f);  // / 128^3
}

// Phase 2: both images resident in LDS; recompute stats + per-channel ssd, form
// mind = exp(-mind/var) for both, accumulate squared difference.
__global__ __launch_bounds__(NTHREADS) void mind_phase2(const float* imgT, const float* imgP,
                                                        const float* mTp, const float* mPp,
                                                        float* partials) {
  __shared__ float simgT[IMZ * IMY * IMX];
  __shared__ float simgP[IMZ * IMY * IMX];
  __shared__ float bufA[DZ * DY * DX];
  __shared__ float bufB[DZ * DY * TX];
  __shared__ float statTmin[NVOX];
  __shared__ float statTmv[NVOX];
  __shared__ float statPmin[NVOX];
  __shared__ float statPmv[NVOX];
  __shared__ float sssd[NVOX];
  __shared__ float sred[NTHREADS];
  const int gx0 = blockIdx.x * TX, gy0 = blockIdx.y * TY, gz0 = blockIdx.z * TZ;

  load_tile(simgT, imgT, gx0, gy0, gz0);
  load_tile(simgP, imgP, gx0, gy0, gz0);

  // stats for T
  for (int i = threadIdx.x; i < NVOX; i += NTHREADS) { statTmin[i] = 3.4e38f; statTmv[i] = 0.f; }
  wait_async_lds();
  for (int c = 0; c < 12; ++c)
    box_channel(simgT, c, gx0, gy0, gz0, bufA, bufB, MODE_STATS,
                statTmin, statTmv, nullptr, nullptr, nullptr, nullptr, nullptr,
                0.f, 0.f, 0.f, 0.f, nullptr);
  __syncthreads();
  for (int i = threadIdx.x; i < NVOX; i += NTHREADS)
    statTmv[i] = statTmv[i] * (1.0f / 12.0f) - statTmin[i];

  // stats for P
  for (int i = threadIdx.x; i < NVOX; i += NTHREADS) { statPmin[i] = 3.4e38f; statPmv[i] = 0.f; }
  for (int c = 0; c < 12; ++c)
    box_channel(simgP, c, gx0, gy0, gz0, bufA, bufB, MODE_STATS,
                statPmin, statPmv, nullptr, nullptr, nullptr, nullptr, nullptr,
                0.f, 0.f, 0.f, 0.f, nullptr);
  __syncthreads();
  for (int i = threadIdx.x; i < NVOX; i += NTHREADS)
    statPmv[i] = statPmv[i] * (1.0f / 12.0f) - statPmin[i];

  const float mT = *mTp, mP = *mPp;
  const float tLo = mT * 0.001f, tHi = mT * 1000.0f;
  const float pLo = mP * 0.001f, pHi = mP * 1000.0f;

  float lossAcc = 0.f;
  for (int c = 0; c < 12; ++c) {
    box_channel(simgT, c, gx0, gy0, gz0, bufA, bufB, MODE_WRITE,
                nullptr, nullptr, sssd, nullptr, nullptr, nullptr, nullptr,
                0.f, 0.f, 0.f, 0.f, nullptr);
    box_channel(simgP, c, gx0, gy0, gz0, bufA, bufB, MODE_LOSS,
                nullptr, nullptr, sssd, statTmin, statTmv, statPmin, statPmv,
                tLo, tHi, pLo, pHi, &lossAcc);
  }
  __syncthreads();
  const int bid = blockIdx.x + gridDim.x * (blockIdx.y + gridDim.y * blockIdx.z);
  block_reduce_store(lossAcc, sred, &partials[bid]);
}

__global__ __launch_bounds__(NTHREADS) void mind_final(const float* partials, float* out) {
  __shared__ float sred[NTHREADS];
  float acc = 0.f;
  for (int i = threadIdx.x; i < NBLOCKS; i += NTHREADS) acc += partials[i];
  sred[threadIdx.x] = acc;
  __syncthreads();
  for (int s = NTHREADS / 2; s > 0; s >>= 1) {
    if (threadIdx.x < (unsigned)s) sred[threadIdx.x] += sred[threadIdx.x + s];
    __syncthreads();
  }
  if (threadIdx.x == 0) out[0] = sred[0] * (1.0f / 25165824.0f);  // / (12*128^3)
}

extern "C" void kernel_launch(void* const* d_in, const int* in_sizes, int n_in,
                              void* d_out, int out_size, void* d_ws, size_t ws_size,
                              hipStream_t stream) {
  const float* yT = (const float*)d_in[0];  // y_true
  const float* yP = (const float*)d_in[1];  // y_pred
  float* out = (float*)d_out;
  float* ws = (float*)d_ws;
  float* part1T = ws;           // [0,1024)
  float* part1P = ws + 1024;    // [1024,2048)
  float* part2  = ws + 2048;    // [2048,3072)
  float* mT     = ws + 3072;
  float* mP     = ws + 3073;

  dim3 grid(N3 / TX, N3 / TY, N3 / TZ);  // 8 x 8 x 16 = 1024 blocks
  dim3 block(NTHREADS);

  mind_phase1<<<grid, block, 0, stream>>>(yT, part1T);
  mind_phase1<<<grid, block, 0, stream>>>(yP, part1P);
  mind_reduce_m<<<dim3(2), block, 0, stream>>>(part1T, part1P, mT, mP);
  mind_phase2<<<grid, block, 0, stream>>>(yT, yP, mT, mP, part2);
  mind_final<<<dim3(1), block, 0, stream>>>(part2, out);
}